// LocalGNNLayer_81784767250586
// MI455X (gfx1250) — compile-verified
//
#include <hip/hip_runtime.h>

// ---------------------------------------------------------------------------
// GATv2 layer for gfx1250 (MI455X), fp32 end-to-end.
// Dims fixed by the reference: IN=OUT=128, HEADS=4, HEAD_DIM=32.
// ---------------------------------------------------------------------------

typedef float v2f __attribute__((ext_vector_type(2)));
typedef float v8f __attribute__((ext_vector_type(8)));

#define NEG_SLOPE 0.2f
#define LN_EPS 1e-5f

// --------------------------- init ------------------------------------------
__global__ __launch_bounds__(256) void k_init(float* __restrict__ out,
                                              unsigned* __restrict__ mkeys,
                                              float* __restrict__ ssum, int N) {
  int i = blockIdx.x * 256 + threadIdx.x;
  if (i < N * 128) out[i] = 0.f;
  if (i < N * 4) { mkeys[i] = 0u; ssum[i] = 0.f; }
}

// --------------------------- dual GEMM (WMMA f32) --------------------------
// XL = x @ Wl, XR = x @ Wr.  Block = 256 threads = 8 waves.
// Block computes a 16-row slab; wave w computes cols [16w, 16w+16) of both
// outputs.  x tile staged in LDS with row stride 132 (conflict-free strided
// A-fragment reads: bank = (4*m + c) % 64, all distinct across the wave).
__global__ __launch_bounds__(256) void k_gemm(const float* __restrict__ x,
                                              const float* __restrict__ Wl,
                                              const float* __restrict__ Wr,
                                              float* __restrict__ XL,
                                              float* __restrict__ XR) {
  __shared__ float tile[16][132];
  const int tid = threadIdx.x;
  const int rowbase = blockIdx.x * 16;

  // cooperative, coalesced load of the 16x128 x tile
  for (int i = tid; i < 16 * 128; i += 256) {
    int r = i >> 7, c = i & 127;
    tile[r][c] = x[(rowbase + r) * 128 + c];
  }
  __syncthreads();

  const int wave = tid >> 5;            // 0..7
  const int lane = tid & 31;
  const int colbase = wave * 16;
  const int mrow = lane & 15;           // A row  (M)
  const int ncol = lane & 15;           // B col  (N)
  const int koff = (lane >> 4) * 2;     // lanes 16-31 hold K=2,3 (ISA 7.12.2)

  v8f accl = {0.f, 0.f, 0.f, 0.f, 0.f, 0.f, 0.f, 0.f};
  v8f accr = {0.f, 0.f, 0.f, 0.f, 0.f, 0.f, 0.f, 0.f};

  for (int kb = 0; kb < 32; ++kb) {     // K = 128 in steps of 4
    const int k0 = kb * 4;
    v2f a;
    a.x = tile[mrow][k0 + koff];
    a.y = tile[mrow][k0 + koff + 1];
    v2f bl, br;
    bl.x = Wl[(k0 + koff) * 128 + colbase + ncol];
    bl.y = Wl[(k0 + koff + 1) * 128 + colbase + ncol];
    br.x = Wr[(k0 + koff) * 128 + colbase + ncol];
    br.y = Wr[(k0 + koff + 1) * 128 + colbase + ncol];
    // (neg_a, A, neg_b, B, c_mod, C, reuse_a, reuse_b)
    accl = __builtin_amdgcn_wmma_f32_16x16x4_f32(false, a, false, bl,
                                                 (short)0, accl, false, false);
    accr = __builtin_amdgcn_wmma_f32_16x16x4_f32(false, a, false, br,
                                                 (short)0, accr, false, false);
  }

  // C/D layout (ISA 7.12.2): VGPR v -> row v (lanes 0-15) / v+8 (lanes 16-31)
  const int rowoff = (lane >> 4) * 8;
  const int c = colbase + (lane & 15);
#pragma unroll
  for (int v = 0; v < 8; ++v) {
    const int r = rowbase + rowoff + v;
    XL[r * 128 + c] = accl[v];
    XR[r * 128 + c] = accr[v];
  }
}

// --------------------------- edge phase helpers ----------------------------
__device__ __forceinline__ unsigned f32_key(float f) {
  unsigned b = __float_as_uint(f);
  return (b & 0x80000000u) ? ~b : (b | 0x80000000u);  // order-preserving
}
__device__ __forceinline__ float key_f32(unsigned k) {
  return __uint_as_float((k & 0x80000000u) ? (k ^ 0x80000000u) : ~k);
}

__device__ __forceinline__ void edge_sd(const int* __restrict__ ei, long e,
                                        int E, int& s, int& d) {
  if (e < E) { s = ei[e]; d = ei[E + e]; }
  else       { s = d = (int)(e - E); }          // self-loop
}

// per-head leakyrelu(xl[s]+xr[d]) . att, wave-reduced. lg[j] valid on all lanes.
__device__ __forceinline__ void edge_logits_calc(const float* __restrict__ XL,
                                                 const float* __restrict__ XR,
                                                 const float* __restrict__ att,
                                                 int s, int d, int lane,
                                                 float lg[4]) {
#pragma unroll
  for (int j = 0; j < 4; ++j) {
    const int c = j * 32 + lane;                 // chunk j == head j
    float h = XL[s * 128 + c] + XR[d * 128 + c];
    h = h > 0.f ? h : NEG_SLOPE * h;
    lg[j] = h * att[c];
  }
#pragma unroll
  for (int off = 16; off > 0; off >>= 1) {
#pragma unroll
    for (int j = 0; j < 4; ++j) lg[j] += __shfl_xor(lg[j], off, 32);
  }
}

// --------------------------- pass A: logits + segment max ------------------
__global__ __launch_bounds__(256) void k_logits(const int* __restrict__ ei,
                                                int E, int ET,
                                                const float* __restrict__ XL,
                                                const float* __restrict__ XR,
                                                const float* __restrict__ att,
                                                float* __restrict__ logits,
                                                unsigned* __restrict__ mkeys) {
  const int lane = threadIdx.x & 31;
  const long e = (long)blockIdx.x * 8 + (threadIdx.x >> 5);
  if (e >= ET) return;
  int s, d;
  edge_sd(ei, e, E, s, d);
  float lg[4];
  edge_logits_calc(XL, XR, att, s, d, lane, lg);
  if (lane < 4) {
    const float v = lane == 0 ? lg[0] : lane == 1 ? lg[1] : lane == 2 ? lg[2] : lg[3];
    if (logits) logits[e * 4 + lane] = v;
    atomicMax(&mkeys[d * 4 + lane], f32_key(v));
  }
}

// --------------------------- pass B: softmax numerator + denominator -------
__device__ __forceinline__ void accum_body(const float* __restrict__ XL,
                                           const unsigned* __restrict__ mkeys,
                                           float* __restrict__ ssum,
                                           float* __restrict__ out,
                                           int s, int d, int lane,
                                           const float lg[4]) {
  float p[4];
#pragma unroll
  for (int j = 0; j < 4; ++j)
    p[j] = __expf(lg[j] - key_f32(mkeys[d * 4 + j]));
  if (lane < 4) {
    const float v = lane == 0 ? p[0] : lane == 1 ? p[1] : lane == 2 ? p[2] : p[3];
    atomicAdd(&ssum[d * 4 + lane], v);
  }
#pragma unroll
  for (int j = 0; j < 4; ++j) {
    const int c = j * 32 + lane;
    atomicAdd(&out[d * 128 + c], p[j] * XL[s * 128 + c]);
  }
}

__global__ __launch_bounds__(256) void k_accum_stored(
    const int* __restrict__ ei, int E, int ET, const float* __restrict__ XL,
    const float* __restrict__ logits, const unsigned* __restrict__ mkeys,
    float* __restrict__ ssum, float* __restrict__ out) {
  const int lane = threadIdx.x & 31;
  const long e = (long)blockIdx.x * 8 + (threadIdx.x >> 5);
  if (e >= ET) return;
  int s, d;
  edge_sd(ei, e, E, s, d);
  float lg[4];
#pragma unroll
  for (int j = 0; j < 4; ++j) lg[j] = logits[e * 4 + j];
  accum_body(XL, mkeys, ssum, out, s, d, lane, lg);
}

__global__ __launch_bounds__(256) void k_accum_recompute(
    const int* __restrict__ ei, int E, int ET, const float* __restrict__ XL,
    const float* __restrict__ XR, const float* __restrict__ att,
    const unsigned* __restrict__ mkeys, float* __restrict__ ssum,
    float* __restrict__ out) {
  const int lane = threadIdx.x & 31;
  const long e = (long)blockIdx.x * 8 + (threadIdx.x >> 5);
  if (e >= ET) return;
  int s, d;
  edge_sd(ei, e, E, s, d);
  float lg[4];
  edge_logits_calc(XL, XR, att, s, d, lane, lg);
  accum_body(XL, mkeys, ssum, out, s, d, lane, lg);
}

// --------------------------- finalize: /s, +bias, ELU, LayerNorm -----------
__global__ __launch_bounds__(256) void k_final(float* __restrict__ out,
                                               const float* __restrict__ ssum,
                                               const float* __restrict__ bias,
                                               const float* __restrict__ gamma,
                                               const float* __restrict__ beta,
                                               int N) {
  const int lane = threadIdx.x & 31;
  const int n = blockIdx.x * 8 + (threadIdx.x >> 5);
  if (n >= N) return;
  float v[4];
  float sum = 0.f, sq = 0.f;
#pragma unroll
  for (int j = 0; j < 4; ++j) {
    const int c = j * 32 + lane;
    float x = out[n * 128 + c] / ssum[n * 4 + j] + bias[c];
    x = x > 0.f ? x : (__expf(x) - 1.f);        // ELU (alpha=1)
    v[j] = x;
    sum += x;
    sq += x * x;
  }
#pragma unroll
  for (int off = 16; off > 0; off >>= 1) {
    sum += __shfl_xor(sum, off, 32);
    sq  += __shfl_xor(sq, off, 32);
  }
  const float mu = sum * (1.f / 128.f);
  const float var = sq * (1.f / 128.f) - mu * mu;
  const float inv = rsqrtf(var + LN_EPS);
#pragma unroll
  for (int j = 0; j < 4; ++j) {
    const int c = j * 32 + lane;
    out[n * 128 + c] = (v[j] - mu) * inv * gamma[c] + beta[c];
  }
}

// ---------------------------------------------------------------------------
extern "C" void kernel_launch(void* const* d_in, const int* in_sizes, int n_in,
                              void* d_out, int out_size, void* d_ws,
                              size_t ws_size, hipStream_t stream) {
  const float* x     = (const float*)d_in[0];
  const int*   ei    = (const int*)d_in[1];
  const float* Wl    = (const float*)d_in[2];
  const float* Wr    = (const float*)d_in[3];
  const float* att   = (const float*)d_in[4];
  const float* bias  = (const float*)d_in[5];
  const float* gamma = (const float*)d_in[6];
  const float* beta  = (const float*)d_in[7];
  float* out = (float*)d_out;

  const int N = in_sizes[0] / 128;
  const int E = in_sizes[1] / 2;
  const int ET = E + N;

  // workspace carve-up (floats)
  float* XL = (float*)d_ws;
  float* XR = XL + (size_t)N * 128;
  size_t base = (size_t)2 * N * 128;                  // floats used so far
  const size_t need_logits = base + (size_t)ET * 4 + (size_t)N * 8;
  const bool have_logits = ws_size >= need_logits * sizeof(float);

  float* logits = have_logits ? (XR + (size_t)N * 128) : nullptr;
  float* tail = have_logits ? (logits + (size_t)ET * 4) : (XR + (size_t)N * 128);
  unsigned* mkeys = (unsigned*)tail;
  float* ssum = (float*)(mkeys + (size_t)N * 4);

  const int edge_blocks = (ET + 7) / 8;

  k_init<<<(N * 128 + 255) / 256, 256, 0, stream>>>(out, mkeys, ssum, N);
  k_gemm<<<(N + 15) / 16, 256, 0, stream>>>(x, Wl, Wr, XL, XR);
  k_logits<<<edge_blocks, 256, 0, stream>>>(ei, E, ET, XL, XR, att, logits, mkeys);
  if (have_logits)
    k_accum_stored<<<edge_blocks, 256, 0, stream>>>(ei, E, ET, XL, logits,
                                                    mkeys, ssum, out);
  else
    k_accum_recompute<<<edge_blocks, 256, 0, stream>>>(ei, E, ET, XL, XR, att,
                                                       mkeys, ssum, out);
  k_final<<<(N + 7) / 8, 256, 0, stream>>>(out, ssum, bias, gamma, beta, N);
}